// MODEL_82781199663612
// MI455X (gfx1250) — compile-verified
//
#include <hip/hip_runtime.h>
#include <cstdint>
#include <cstddef>

// ---------------------------------------------------------------------------
// DKVMN-style knowledge tracing forward pass for gfx1250 (MI455X).
// Dense linears run on V_WMMA_F32_16X16X4_F32 (exact fp32, matrix pipe).
// GEMM staging uses GLOBAL_LOAD_ASYNC_TO_LDS_B128 + LDS double buffering
// (ASYNCcnt path), overlapping the next K-tile's global->LDS DMA with the
// current tile's WMMAs.  The recurrent memory scan keeps each Mv column in
// registers (no barriers).
// ---------------------------------------------------------------------------

typedef float v2f __attribute__((ext_vector_type(2)));
typedef float v8f __attribute__((ext_vector_type(8)));
typedef int   v4i_vs __attribute__((vector_size(16)));   // type expected by async-LDS builtin

#if defined(__has_builtin)
#if __has_builtin(__builtin_amdgcn_global_load_async_to_lds_b128) && \
    __has_builtin(__builtin_amdgcn_s_wait_asynccnt)
#define USE_ASYNC_LDS 1
#endif
#endif
#ifndef USE_ASYNC_LDS
#define USE_ASYNC_LDS 0
#endif

static constexpr int    Bn  = 256;
static constexpr int    Sn  = 500;
static constexpr size_t Nn  = (size_t)Bn * Sn;   // 128000 rows
static constexpr int    DQ  = 128;
static constexpr int    DV  = 256;
static constexpr int    Mn  = 50;
static constexpr int    Fn  = 512;
static constexpr int    KPI = DV + DQ;           // 384

// -------------------- workspace layout (floats) ----------------------------
static constexpr size_t off_PI   = 0;                       // [N,384] predict_input (reads | input_embed)
static constexpr size_t off_E    = off_PI + Nn * KPI;       // [N,256] erase gates
static constexpr size_t off_A    = off_E  + Nn * DV;        // [N,256] add vectors
static constexpr size_t off_W    = off_A  + Nn * DV;        // [N,50]  attention weights
static constexpr size_t off_R0   = off_W  + Nn * Mn;        // te
static constexpr size_t off_R1   = off_R0 + Nn * DQ;        // qe -> QP
static constexpr size_t off_R2   = off_R1 + Nn * DQ;        // sum_h
static constexpr size_t off_R3   = off_R2 + Nn * DQ;        // X1 -> TF
static constexpr size_t off_R4   = off_R3 + Nn * DQ;        // X2 -> DF
static constexpr size_t off_R5   = off_R4 + Nn * DQ;        // H temp -> X3
static constexpr size_t off_pred = off_R5 + Nn * DQ;        // [N] fused pred accum
static constexpr size_t off_acc  = off_pred + Nn;           // [2] loss sum / count
// qa_emb gather aliases off_PI (dead before any PI write).

// -------------------- small utility kernels --------------------------------
__global__ void zero_kernel(float* __restrict__ p, int n) {
  int i = blockIdx.x * 256 + threadIdx.x;
  if (i < n) p[i] = 0.0f;
}

// Gather DQ-wide embeddings and form the input sums.
__global__ __launch_bounds__(256) void gather_dq_kernel(
    const int* __restrict__ q_data, const int* __restrict__ time_data,
    const int* __restrict__ attempt_data, const int* __restrict__ hint_data,
    const int* __restrict__ hintTotal_data,
    const float* __restrict__ q_emb, const float* __restrict__ time_emb,
    const float* __restrict__ attempt_emb, const float* __restrict__ ht_emb,
    float* __restrict__ te_o, float* __restrict__ qe_o,
    float* __restrict__ sumh_o, float* __restrict__ x1_o, float* __restrict__ x2_o) {
  size_t idx = (size_t)blockIdx.x * 256 + threadIdx.x;   // N*128 threads
  size_t row = idx >> 7;
  int    d   = (int)(idx & 127);
  float qe  = q_emb[(size_t)q_data[row] * DQ + d];
  float te  = time_emb[(size_t)time_data[row] * DQ + d];
  float ae  = attempt_emb[(size_t)attempt_data[row] * DQ + d];
  float he  = ht_emb[(size_t)hint_data[row] * DQ + d];
  float hte = ht_emb[(size_t)hintTotal_data[row] * DQ + d];
  float sh  = he + hte + ae;
  te_o[idx]   = te;
  qe_o[idx]   = qe;
  sumh_o[idx] = sh;
  x1_o[idx]   = te + qe;
  x2_o[idx]   = qe + sh;
}

__global__ __launch_bounds__(256) void gather_qa_kernel(
    const int* __restrict__ qa_data, const float* __restrict__ qa_emb,
    float* __restrict__ qa_o) {
  size_t idx = (size_t)blockIdx.x * 256 + threadIdx.x;   // N*256 threads
  size_t row = idx >> 8;
  int    d   = (int)(idx & 255);
  qa_o[idx] = qa_emb[(size_t)qa_data[row] * DV + d];
}

// X3 = qe + DF * sum_h
__global__ __launch_bounds__(256) void ew_x3_kernel(
    const float* __restrict__ qe, const float* __restrict__ df,
    const float* __restrict__ sumh, float* __restrict__ x3) {
  size_t idx = (size_t)blockIdx.x * 256 + threadIdx.x;
  x3[idx] = fmaf(df[idx], sumh[idx], qe[idx]);
}

// predict_input[:, 256:384] = QP + TF * te
__global__ __launch_bounds__(256) void ew_ie_kernel(
    const float* __restrict__ qp, const float* __restrict__ tf,
    const float* __restrict__ te, float* __restrict__ pi) {
  size_t idx = (size_t)blockIdx.x * 256 + threadIdx.x;
  size_t row = idx >> 7;
  int    d   = (int)(idx & 127);
  pi[row * KPI + DV + d] = fmaf(tf[idx], te[idx], qp[idx]);
}

// -------------------- WMMA fp32 GEMM: C = act(A @ W^T + b) ------------------
// A: [Nrows,K] row-major, W: [O,K] row-major (torch Linear), C: [Nrows,O].
// Block tile 64x64 (8 waves: 4 along M, 2 along N; 16x32 per wave).
// ACT: 0=none 1=tanh 2=sigmoid.  FUSE: accumulate pred += act()*predW[col].
static constexpr int BM  = 64;
static constexpr int BN  = 64;
static constexpr int BK  = 32;
static constexpr int LDP = 36;   // padded LDS row stride (conflict-free b64 reads)
#if USE_ASYNC_LDS
static constexpr int NBUF = 2;   // LDS double buffer for async staging
#else
static constexpr int NBUF = 1;
#endif

template <int ACT, bool FUSE>
__global__ __launch_bounds__(256) void gemm_wmma_kernel(
    const float* __restrict__ A, const float* __restrict__ W,
    const float* __restrict__ bias, float* __restrict__ C,
    const float* __restrict__ predW, float* __restrict__ pred,
    int K, int O) {
  __shared__ float sA[NBUF][BM * LDP];
  __shared__ float sW[NBUF][BN * LDP];
  __shared__ float sPred[BM];

  const int tid   = threadIdx.x;
  const int wid   = tid >> 5;
  const int lane  = tid & 31;
  const int lhalf = lane & 15;
  const int hi    = (lane >= 16) ? 1 : 0;
  const int wm    = wid & 3;   // M sub-tile: wm*16
  const int wn    = wid >> 2;  // N sub-tile: wn*32 (two 16-wide fragments)
  const size_t m0 = (size_t)blockIdx.y * BM;
  const int    o0 = blockIdx.x * BN;

  // Per-thread staging assignment: two 16B chunks of each tile.
  const int row0 = tid >> 3;           // 0..31
  const int row1 = row0 + 32;          // 32..63
  const int kk0  = (tid & 7) << 2;     // 0,4,...,28

  if (FUSE && tid < BM) sPred[tid] = 0.0f;

  v8f c0 = {};
  v8f c1 = {};

#if USE_ASYNC_LDS
  // ---- async staging: GLOBAL_LOAD_ASYNC_TO_LDS_B128, ASYNCcnt-tracked ----
  auto issue_tile = [&](int kb, int buf) {
    __builtin_amdgcn_global_load_async_to_lds_b128(
        (v4i_vs*)(const_cast<float*>(A) + (m0 + row0) * K + kb + kk0),
        (v4i_vs*)&sA[buf][row0 * LDP + kk0], 0, 0);
    __builtin_amdgcn_global_load_async_to_lds_b128(
        (v4i_vs*)(const_cast<float*>(A) + (m0 + row1) * K + kb + kk0),
        (v4i_vs*)&sA[buf][row1 * LDP + kk0], 0, 0);
    __builtin_amdgcn_global_load_async_to_lds_b128(
        (v4i_vs*)(const_cast<float*>(W) + (size_t)(o0 + row0) * K + kb + kk0),
        (v4i_vs*)&sW[buf][row0 * LDP + kk0], 0, 0);
    __builtin_amdgcn_global_load_async_to_lds_b128(
        (v4i_vs*)(const_cast<float*>(W) + (size_t)(o0 + row1) * K + kb + kk0),
        (v4i_vs*)&sW[buf][row1 * LDP + kk0], 0, 0);
  };
  issue_tile(0, 0);
  int buf = 0;
  for (int kb = 0; kb < K; kb += BK) {
    __builtin_amdgcn_s_wait_asynccnt(0);   // this wave's tile landed in LDS
    __syncthreads();                       // every wave's tile landed
    if (kb + BK < K) issue_tile(kb + BK, buf ^ 1);  // overlaps WMMAs below
    const float* tA = sA[buf];
    const float* tW = sW[buf];
#pragma unroll
    for (int k4 = 0; k4 < BK; k4 += 4) {
      const int kl = k4 + (hi ? 2 : 0);
      v2f a  = *reinterpret_cast<const v2f*>(&tA[(wm * 16 + lhalf) * LDP + kl]);
      v2f b0 = *reinterpret_cast<const v2f*>(&tW[(wn * 32 + lhalf) * LDP + kl]);
      v2f b1 = *reinterpret_cast<const v2f*>(&tW[(wn * 32 + 16 + lhalf) * LDP + kl]);
      c0 = __builtin_amdgcn_wmma_f32_16x16x4_f32(false, a, false, b0, (short)0, c0, false, false);
      c1 = __builtin_amdgcn_wmma_f32_16x16x4_f32(false, a, false, b1, (short)0, c1, false, false);
    }
    __syncthreads();                       // buf reusable for tile kb+2*BK
    buf ^= 1;
  }
#else
  // ---- fallback: register-prefetch software pipeline ----
  float4 pa0 = *reinterpret_cast<const float4*>(A + (m0 + row0) * K + kk0);
  float4 pa1 = *reinterpret_cast<const float4*>(A + (m0 + row1) * K + kk0);
  float4 pw0 = *reinterpret_cast<const float4*>(W + (size_t)(o0 + row0) * K + kk0);
  float4 pw1 = *reinterpret_cast<const float4*>(W + (size_t)(o0 + row1) * K + kk0);
  for (int kb = 0; kb < K; kb += BK) {
    *reinterpret_cast<float4*>(&sA[0][row0 * LDP + kk0]) = pa0;
    *reinterpret_cast<float4*>(&sA[0][row1 * LDP + kk0]) = pa1;
    *reinterpret_cast<float4*>(&sW[0][row0 * LDP + kk0]) = pw0;
    *reinterpret_cast<float4*>(&sW[0][row1 * LDP + kk0]) = pw1;
    __syncthreads();
    if (kb + BK < K) {  // issue next tile's loads; overlap with WMMAs below
      pa0 = *reinterpret_cast<const float4*>(A + (m0 + row0) * K + kb + BK + kk0);
      pa1 = *reinterpret_cast<const float4*>(A + (m0 + row1) * K + kb + BK + kk0);
      pw0 = *reinterpret_cast<const float4*>(W + (size_t)(o0 + row0) * K + kb + BK + kk0);
      pw1 = *reinterpret_cast<const float4*>(W + (size_t)(o0 + row1) * K + kb + BK + kk0);
    }
#pragma unroll
    for (int k4 = 0; k4 < BK; k4 += 4) {
      const int kl = k4 + (hi ? 2 : 0);
      v2f a  = *reinterpret_cast<const v2f*>(&sA[0][(wm * 16 + lhalf) * LDP + kl]);
      v2f b0 = *reinterpret_cast<const v2f*>(&sW[0][(wn * 32 + lhalf) * LDP + kl]);
      v2f b1 = *reinterpret_cast<const v2f*>(&sW[0][(wn * 32 + 16 + lhalf) * LDP + kl]);
      c0 = __builtin_amdgcn_wmma_f32_16x16x4_f32(false, a, false, b0, (short)0, c0, false, false);
      c1 = __builtin_amdgcn_wmma_f32_16x16x4_f32(false, a, false, b1, (short)0, c1, false, false);
    }
    __syncthreads();
  }
#endif

  // Epilogue.  D layout: reg i, lanes 0-15 -> row i, col lane; lanes 16-31 -> row 8+i.
  const int rloc = wm * 16 + (hi ? 8 : 0);
#pragma unroll
  for (int f = 0; f < 2; ++f) {
    v8f& c = f ? c1 : c0;
    const int cg = o0 + wn * 32 + f * 16 + lhalf;
    const float bv = bias[cg];
    const float pw = FUSE ? predW[cg] : 0.0f;
#pragma unroll
    for (int i = 0; i < 8; ++i) {
      float v = c[i] + bv;
      if (ACT == 1) v = tanhf(v);
      else if (ACT == 2) v = 1.0f / (1.0f + expf(-v));
      if (FUSE) {
        atomicAdd(&sPred[rloc + i], v * pw);          // ds_add_f32
      } else {
        C[(m0 + rloc + i) * O + cg] = v;
      }
    }
  }
  if (FUSE) {
    __syncthreads();
    if (tid < BM) atomicAdd(&pred[m0 + tid], sPred[tid]);  // global_atomic_add_f32
  }
}

// -------------------- attention softmax: w = softmax(QP @ Mk^T) -------------
__global__ __launch_bounds__(256) void wsoftmax_kernel(
    const float* __restrict__ QP, const float* __restrict__ Mk,
    float* __restrict__ Wout) {
  __shared__ float sl[4 * 64];
  const int rl  = threadIdx.x >> 6;     // 4 rows per block
  const int m   = threadIdx.x & 63;
  const size_t row = (size_t)blockIdx.x * 4 + rl;
  float logit = 0.0f;
  if (m < Mn) {
    const float* q = QP + row * DQ;
    const float* k = Mk + (size_t)m * DQ;
#pragma unroll 8
    for (int kk = 0; kk < DQ; ++kk) logit = fmaf(q[kk], k[kk], logit);
    sl[rl * 64 + m] = logit;
  }
  __syncthreads();
  float mx = -1e30f;
  if (m < Mn)
    for (int j = 0; j < Mn; ++j) mx = fmaxf(mx, sl[rl * 64 + j]);
  __syncthreads();
  float ex = 0.0f;
  if (m < Mn) { ex = expf(logit - mx); sl[rl * 64 + m] = ex; }
  __syncthreads();
  if (m < Mn) {
    float sum = 0.0f;
    for (int j = 0; j < Mn; ++j) sum += sl[rl * 64 + j];
    Wout[row * Mn + m] = ex / sum;
  }
}

// -------------------- recurrent memory scan ---------------------------------
// One block per batch element; thread d owns column d of Mv across all 50
// slots, kept in registers for the full 500-step recurrence (no barriers).
__global__ __launch_bounds__(256) void scan_kernel(
    const float* __restrict__ wbuf, const float* __restrict__ Ebuf,
    const float* __restrict__ Abuf, const float* __restrict__ Mv0,
    float* __restrict__ PI) {
  const int b = blockIdx.x;
  const int d = threadIdx.x;          // 0..255 == DV
  float col[Mn];
#pragma unroll
  for (int m = 0; m < Mn; ++m) col[m] = Mv0[(size_t)m * DV + d];

  for (int s = 0; s < Sn; ++s) {
    const size_t row = (size_t)b * Sn + s;
    const float e = Ebuf[row * DV + d];
    const float a = Abuf[row * DV + d];
    const float* wr = wbuf + row * Mn;  // block-uniform -> scalar loads
    float rd = 0.0f;
#pragma unroll
    for (int m = 0; m < Mn; ++m) {
      const float wm  = wr[m];
      const float old = col[m];
      rd = fmaf(wm, old, rd);                      // read uses pre-update Mv
      col[m] = fmaf(wm, a, old - old * (wm * e));  // Mv*(1-w*e) + w*a
    }
    PI[row * KPI + d] = rd;             // reads -> predict_input[:, :256]
  }
}

// -------------------- masked BCE loss ---------------------------------------
__global__ __launch_bounds__(256) void loss_partial_kernel(
    const float* __restrict__ pred, const float* __restrict__ pred_b,
    const int* __restrict__ target, float* __restrict__ out,
    float* __restrict__ acc, int N) {
  __shared__ float ssum[256];
  __shared__ float scnt[256];
  const int n = blockIdx.x * 256 + threadIdx.x;
  float l = 0.0f, cf = 0.0f;
  if (n < N) {
    const float p = pred[n] + pred_b[0];
    const int   t = target[n];
    const bool  msk = (t >= 1);
    const float y = msk ? (float)(t - 1) : 0.0f;
    const float sp = (p > 0.0f) ? (p + log1pf(expf(-p))) : log1pf(expf(p));
    const float sg = 1.0f / (1.0f + expf(-p));
    out[1 + n]     = msk ? sg : 0.0f;
    out[1 + N + n] = msk ? y : 0.0f;
    if (msk) { l = sp - p * y; cf = 1.0f; }
  }
  ssum[threadIdx.x] = l;
  scnt[threadIdx.x] = cf;
  __syncthreads();
  for (int s = 128; s > 0; s >>= 1) {
    if (threadIdx.x < s) {
      ssum[threadIdx.x] += ssum[threadIdx.x + s];
      scnt[threadIdx.x] += scnt[threadIdx.x + s];
    }
    __syncthreads();
  }
  if (threadIdx.x == 0) {
    atomicAdd(&acc[0], ssum[0]);
    atomicAdd(&acc[1], scnt[0]);
  }
}

__global__ void loss_final_kernel(const float* __restrict__ acc, float* __restrict__ out) {
  out[0] = acc[0] / fmaxf(acc[1], 1.0f);
}

// ---------------------------------------------------------------------------
extern "C" void kernel_launch(void* const* d_in, const int* in_sizes, int n_in,
                              void* d_out, int out_size, void* d_ws, size_t ws_size,
                              hipStream_t stream) {
  (void)in_sizes; (void)n_in; (void)out_size; (void)ws_size;

  const int*   q_data    = (const int*)d_in[0];
  const int*   qa_data   = (const int*)d_in[1];
  const int*   target    = (const int*)d_in[2];
  const int*   time_data = (const int*)d_in[3];
  const int*   att_data  = (const int*)d_in[4];
  const int*   hint_data = (const int*)d_in[5];
  const int*   hintT     = (const int*)d_in[6];
  const float* q_emb     = (const float*)d_in[7];
  const float* qa_emb    = (const float*)d_in[8];
  const float* time_emb  = (const float*)d_in[9];
  const float* att_emb   = (const float*)d_in[10];
  const float* ht_emb    = (const float*)d_in[11];
  const float* Mk        = (const float*)d_in[12];
  const float* Mv0       = (const float*)d_in[13];
  const float* diff_W    = (const float*)d_in[14];
  const float* diff_b    = (const float*)d_in[15];
  const float* diff2_W   = (const float*)d_in[16];
  const float* diff2_b   = (const float*)d_in[17];
  const float* erase_W   = (const float*)d_in[18];
  const float* erase_b   = (const float*)d_in[19];
  const float* add_W     = (const float*)d_in[20];
  const float* add_b     = (const float*)d_in[21];
  const float* read_W    = (const float*)d_in[22];
  const float* read_b    = (const float*)d_in[23];
  const float* pred_W    = (const float*)d_in[24];
  const float* pred_b    = (const float*)d_in[25];
  float*       out       = (float*)d_out;

  float* ws   = (float*)d_ws;
  float* PI   = ws + off_PI;
  float* QAe  = ws + off_PI;        // alias: dead before any PI write
  float* Ebuf = ws + off_E;
  float* Abuf = ws + off_A;
  float* Wbuf = ws + off_W;
  float* R0   = ws + off_R0;        // te
  float* R1   = ws + off_R1;        // qe -> QP
  float* R2   = ws + off_R2;        // sum_h
  float* R3   = ws + off_R3;        // X1 -> TF
  float* R4   = ws + off_R4;        // X2 -> DF
  float* R5   = ws + off_R5;        // H -> X3
  float* pred = ws + off_pred;
  float* acc  = ws + off_acc;

  const dim3 blk(256);
  const dim3 g_dq((unsigned)(Nn * DQ / 256));
  const dim3 g_dv((unsigned)(Nn * DV / 256));
  const dim3 g128(DQ / BN, (unsigned)(Nn / BM));
  const dim3 g256(DV / BN, (unsigned)(Nn / BM));
  const dim3 g512(Fn / BN, (unsigned)(Nn / BM));

  // 0) zero fused-pred accumulator + loss accumulators (graph-replay safe)
  zero_kernel<<<dim3((unsigned)((Nn + 2 + 255) / 256)), blk, 0, stream>>>(pred, (int)Nn + 2);

  // 1) gathers
  gather_dq_kernel<<<g_dq, blk, 0, stream>>>(q_data, time_data, att_data, hint_data, hintT,
                                             q_emb, time_emb, att_emb, ht_emb,
                                             R0, R1, R2, R3, R4);
  gather_qa_kernel<<<g_dv, blk, 0, stream>>>(qa_data, qa_emb, QAe);

  // 2) difficulty / time heads (shared lin1/lin2 applied twice)
  gemm_wmma_kernel<1, false><<<g128, blk, 0, stream>>>(R3, diff_W, diff_b, R5, nullptr, nullptr, DQ, DQ);   // H1 = tanh(lin1(te+qe))
  gemm_wmma_kernel<2, false><<<g128, blk, 0, stream>>>(R5, diff2_W, diff2_b, R3, nullptr, nullptr, DQ, DQ); // TF = sigmoid(lin2(H1))
  gemm_wmma_kernel<1, false><<<g128, blk, 0, stream>>>(R4, diff_W, diff_b, R5, nullptr, nullptr, DQ, DQ);   // H2 = tanh(lin1(hte+he+ae+qe))
  gemm_wmma_kernel<2, false><<<g128, blk, 0, stream>>>(R5, diff2_W, diff2_b, R4, nullptr, nullptr, DQ, DQ); // DF = sigmoid(lin2(H2))
  ew_x3_kernel<<<g_dq, blk, 0, stream>>>(R1, R4, R2, R5);                                                   // X3 = qe + DF*sum_h
  gemm_wmma_kernel<0, false><<<g128, blk, 0, stream>>>(R5, diff_W, diff_b, R1, nullptr, nullptr, DQ, DQ);   // QP = lin1(X3)

  // 3) erase/add gates (must finish before PI writes: QAe aliases PI)
  gemm_wmma_kernel<2, false><<<g256, blk, 0, stream>>>(QAe, erase_W, erase_b, Ebuf, nullptr, nullptr, DV, DV);
  gemm_wmma_kernel<1, false><<<g256, blk, 0, stream>>>(QAe, add_W, add_b, Abuf, nullptr, nullptr, DV, DV);

  // 4) input_embed into predict_input[:, 256:384]
  ew_ie_kernel<<<g_dq, blk, 0, stream>>>(R1, R3, R0, PI);

  // 5) attention weights + recurrent memory scan (reads -> PI[:, :256])
  wsoftmax_kernel<<<dim3((unsigned)(Nn / 4)), blk, 0, stream>>>(R1, Mk, Wbuf);
  scan_kernel<<<dim3(Bn), blk, 0, stream>>>(Wbuf, Ebuf, Abuf, Mv0, PI);

  // 6) readout GEMM with fused pred dot-product (RC never materialized)
  gemm_wmma_kernel<1, true><<<g512, blk, 0, stream>>>(PI, read_W, read_b, nullptr, pred_W, pred, KPI, Fn);

  // 7) masked BCE loss + outputs
  loss_partial_kernel<<<dim3((unsigned)((Nn + 255) / 256)), blk, 0, stream>>>(pred, pred_b, target, out, acc, (int)Nn);
  loss_final_kernel<<<dim3(1), dim3(1), 0, stream>>>(acc, out);
}